// VectorQuantizer_28595892257691
// MI455X (gfx1250) — compile-verified
//
#include <hip/hip_runtime.h>
#include <hip/hip_bf16.h>
#include <math.h>

// ---------------------------------------------------------------------------
// VQ-VAE VectorQuantizer forward for MI455X (gfx1250, wave32, WMMA)
//   N = 32*4096 = 131072 input vectors, D = 64, K = 512 codewords.
//   Distance GEMM done with V_WMMA_F32_16X16X4_F32 (exact fp32, matches ref).
// d_out layout (floats, reference return order, concatenated flat):
//   [0]                 loss
//   [1 .. 1+N*D)        quantized_st   (== gathered codewords)
//   [1+N*D]             perplexity
//   [2+N*D .. 2+N*D+N)  encoding indices (as float)
// ---------------------------------------------------------------------------

typedef float v2f __attribute__((ext_vector_type(2)));
typedef float v8f __attribute__((ext_vector_type(8)));

#define NUM_EMB   512
#define DIM       64
#define N_TOTAL   (32 * 4096)
#define COMMIT    0.25f

// ---------------- init: codebook squared norms, zero accumulators ----------
__global__ void vq_init_kernel(const float* __restrict__ w,
                               float* __restrict__ enorm,
                               float* __restrict__ counts,
                               float* __restrict__ lossAcc) {
    int t = threadIdx.x;            // one block of 512
    if (t < NUM_EMB) {
        const float* row = w + t * DIM;
        float s = 0.f;
#pragma unroll
        for (int c = 0; c < DIM; ++c) { float v = row[c]; s += v * v; }
        enorm[t]  = s;
        counts[t] = 0.f;
    }
    if (t == 0) lossAcc[0] = 0.f;
}

// ---------------- main: WMMA distances + argmin + gather + loss ------------
__launch_bounds__(256)
__global__ void vq_main_kernel(const float* __restrict__ x,
                               const float* __restrict__ w,
                               const float* __restrict__ enorm,
                               float* __restrict__ counts,
                               float* __restrict__ lossAcc,
                               float* __restrict__ q_out,
                               float* __restrict__ idx_out) {
    __shared__ int s_idx[8][16];                 // winner index per row, per wave

    const int lane = threadIdx.x & 31;
    const int wave = threadIdx.x >> 5;
    const int half = lane >> 4;                  // 0: K even pair, 1: K odd pair
    const int l15  = lane & 15;

    const long long rowBase = (long long)blockIdx.x * 128 + (long long)wave * 16;
    const float* xt = x + rowBase * DIM;

    // A-operand (16x4 f32 layout): a[kk] holds x[row=l15][4kk+2*half .. +1]
    v2f a[16];
#pragma unroll
    for (int kk = 0; kk < 16; ++kk) {
        const float* p = xt + l15 * DIM + kk * 4 + 2 * half;
        a[kk].x = p[0];
        a[kk].y = p[1];
    }

    // Running argmin over score = ||e||^2 - 2 x.e   (||x||^2 is row-constant)
    float best[8];
    int   bidx[8];
#pragma unroll
    for (int j = 0; j < 8; ++j) { best[j] = 3.4e38f; bidx[j] = 0; }

    for (int ct = 0; ct < NUM_EMB / 16; ++ct) {
        const int    code = ct * 16 + l15;       // this lane's column (codeword)
        const float* wr   = w + code * DIM + 2 * half;

        v8f acc = {};
#pragma unroll
        for (int kk = 0; kk < 16; ++kk) {        // D = 64 = 16 * (K=4)
            v2f b;
            b.x = wr[kk * 4];
            b.y = wr[kk * 4 + 1];
            acc = __builtin_amdgcn_wmma_f32_16x16x4_f32(
                      false, a[kk], false, b, (short)0, acc, false, false);
        }

        const float en = enorm[code];
#pragma unroll
        for (int j = 0; j < 8; ++j) {
            float d = en - 2.0f * acc[j];        // acc[j]: row j+8*half, col=code
            if (d < best[j]) { best[j] = d; bidx[j] = code; }
        }
    }

    // Argmin across the 16 lanes (columns) sharing each row; prefer lower index
    // on ties to match jnp.argmin's first-occurrence rule.
#pragma unroll
    for (int m = 1; m < 16; m <<= 1) {
#pragma unroll
        for (int j = 0; j < 8; ++j) {
            float s2 = __shfl_xor(best[j], m, 32);
            int   i2 = __shfl_xor(bidx[j], m, 32);
            if (s2 < best[j] || (s2 == best[j] && i2 < bidx[j])) {
                best[j] = s2; bidx[j] = i2;
            }
        }
    }

    if (l15 == 0) {                              // lanes 0 and 16 publish rows
#pragma unroll
        for (int j = 0; j < 8; ++j) {
            int row = j + 8 * half;
            s_idx[wave][row] = bidx[j];
            idx_out[rowBase + row] = (float)bidx[j];
            atomicAdd(&counts[bidx[j]], 1.0f);
        }
    }
    __syncthreads();

    // Gather winning codeword, emit quantized output, accumulate (x-q)^2.
    const int    myIdx = s_idx[wave][l15];
    const float* wq    = w + myIdx * DIM;
    float lsum = 0.f;
#pragma unroll
    for (int kk = 0; kk < 16; ++kk) {
        int   c  = kk * 4 + 2 * half;
        float q0 = wq[c], q1 = wq[c + 1];
        float d0 = a[kk].x - q0;
        float d1 = a[kk].y - q1;
        lsum += d0 * d0 + d1 * d1;
        float* o = q_out + (rowBase + l15) * DIM + c;
        o[0] = q0;
        o[1] = q1;
    }
#pragma unroll
    for (int m = 16; m >= 1; m >>= 1) lsum += __shfl_xor(lsum, m, 32);
    if (lane == 0) atomicAdd(lossAcc, lsum);
}

// ---------------- finalize: loss scalar + perplexity -----------------------
__global__ void vq_finalize_kernel(const float* __restrict__ counts,
                                   const float* __restrict__ lossAcc,
                                   float* __restrict__ loss_out,
                                   float* __restrict__ perp_out) {
    __shared__ float red[NUM_EMB];
    int t = threadIdx.x;                         // one block of 512
    float p = counts[t] * (1.0f / (float)N_TOTAL);
    red[t] = p * logf(p + 1e-10f);
    __syncthreads();
    for (int s = NUM_EMB / 2; s >= 1; s >>= 1) {
        if (t < s) red[t] += red[t + s];
        __syncthreads();
    }
    if (t == 0) {
        perp_out[0] = expf(-red[0]);
        // forward: q_latent_loss == e_latent_loss == mean((x-q)^2)
        loss_out[0] = (1.0f + COMMIT) * lossAcc[0] / (float)((long long)N_TOTAL * DIM);
    }
}

// ---------------------------------------------------------------------------
extern "C" void kernel_launch(void* const* d_in, const int* in_sizes, int n_in,
                              void* d_out, int out_size, void* d_ws, size_t ws_size,
                              hipStream_t stream) {
    const float* x = (const float*)d_in[0];      // [32,4096,64] fp32
    const float* w = (const float*)d_in[1];      // [512,64]     fp32
    // d_in[2] = is_training: forward pass is identical either way.

    float* out      = (float*)d_out;
    float* loss_out = out;                                   // [0]
    float* q_out    = out + 1;                               // [1 .. 1+N*D)
    float* perp_out = out + 1 + (size_t)N_TOTAL * DIM;       // [1+N*D]
    float* idx_out  = perp_out + 1;                          // N floats

    float* enorm   = (float*)d_ws;               // 512 floats
    float* counts  = enorm + NUM_EMB;            // 512 floats
    float* lossAcc = counts + NUM_EMB;           // 1 float

    vq_init_kernel<<<1, 512, 0, stream>>>(w, enorm, counts, lossAcc);
    vq_main_kernel<<<N_TOTAL / 128, 256, 0, stream>>>(x, w, enorm, counts,
                                                      lossAcc, q_out, idx_out);
    vq_finalize_kernel<<<1, 512, 0, stream>>>(counts, lossAcc, loss_out, perp_out);
}